// TypedPrefixCompiler_23338852287192
// MI455X (gfx1250) — compile-verified
//
#include <hip/hip_runtime.h>
#include <hip/hip_bf16.h>
#include <math.h>

typedef __attribute__((ext_vector_type(2))) float v2f;
typedef __attribute__((ext_vector_type(4))) float v4f;
typedef __attribute__((ext_vector_type(8))) float v8f;

#define NBATCH 4
#define NT     8192
#define ND     1024
#define NB     64        // BLOCKS
#define NL     128       // T / BLOCKS
#define NTOP   16
#define NS     21        // TOPK + MACRO + 1
#define NF     2048      // 2*D
#define NQ     64        // SLOTS
#define QSPLIT 8
#define ROWS_PER_SPLIT (NL / QSPLIT)   // 16
#define KW     4         // K-split waves per GEMM workgroup
#define RMS_EPS 1.1920929e-07f

static __device__ __forceinline__ v8f wmma4(v2f a, v2f b, v8f c) {
  // V_WMMA_F32_16X16X4_F32 : D = A(16x4) * B(4x16) + C(16x16), all f32
  return __builtin_amdgcn_wmma_f32_16x16x4_f32(
      /*neg_a=*/false, a, /*neg_b=*/false, b,
      /*c_mod=*/(short)0, c, /*reuse_a=*/false, /*reuse_b=*/false);
}

// ---------------------------------------------------------------------------
// Phase 1a: partial segment sums. grid = B*NB*QSPLIT (2048), block = 256.
// Each group sums 16 rows x 1024 dims (float4 per thread per row).
// ---------------------------------------------------------------------------
__global__ void k_seg_partial(const float* __restrict__ ph, float* __restrict__ part) {
  int g   = blockIdx.x;
  int q   = g & (QSPLIT - 1);
  int seg = g >> 3;                       // b*NB + blk
  int t   = threadIdx.x;
  const float* base = ph + ((size_t)seg * NL + (size_t)q * ROWS_PER_SPLIT) * ND + t * 4;
  v4f acc = {0.f, 0.f, 0.f, 0.f};
#pragma unroll
  for (int l = 0; l < ROWS_PER_SPLIT; ++l)
    acc += *(const v4f*)(base + (size_t)l * ND);
  *(v4f*)(part + (size_t)g * ND + t * 4) = acc;
}

// ---------------------------------------------------------------------------
// Phase 1b: combine partials + NLL block sums. grid = B*NB (256), block = 256.
// ---------------------------------------------------------------------------
__global__ void k_seg_combine(const float* __restrict__ part, const float* __restrict__ nll,
                              float* __restrict__ seg_sum, float* __restrict__ nll_sum) {
  int seg = blockIdx.x;
  int t   = threadIdx.x;
  v4f acc = {0.f, 0.f, 0.f, 0.f};
#pragma unroll
  for (int q = 0; q < QSPLIT; ++q)
    acc += *(const v4f*)(part + ((size_t)seg * QSPLIT + q) * ND + t * 4);
  *(v4f*)(seg_sum + (size_t)seg * ND + t * 4) = acc;

  __shared__ float sn[128];
  int b = seg >> 6, blk = seg & 63;
  if (t < 128) sn[t] = nll[(size_t)b * NT + blk * NL + t];
  __syncthreads();
  for (int off = 64; off > 0; off >>= 1) {
    if (t < off) sn[t] += sn[t + off];
    __syncthreads();
  }
  if (t == 0) nll_sum[seg] = sn[0];
}

// ---------------------------------------------------------------------------
// Phase 2: scores, z-score, top-16 (rank-select, sorted ascending). grid = B.
// ---------------------------------------------------------------------------
__global__ void k_score_topk(const float* __restrict__ seg_sum,
                             const float* __restrict__ nll_sum,
                             int* __restrict__ idx) {
  int b = blockIdx.x, t = threadIdx.x;   // 256 threads
  __shared__ float partq[256];
  __shared__ float hid[NB], sur[NB], sc[NB];
  __shared__ int   sel[NB];
  __shared__ float stats[4];             // mh, sdh, ms, sds

  int i = t >> 2, p = t & 3;
  const float* row = seg_sum + ((size_t)(b * NB + i)) * ND + p * 256;
  float ss = 0.f;
  for (int d = 0; d < 256; ++d) { float m = row[d] * (1.f / NL); ss += m * m; }
  partq[t] = ss;
  __syncthreads();
  if (p == 0) {
    hid[i] = sqrtf(partq[t] + partq[t + 1] + partq[t + 2] + partq[t + 3]);
    sur[i] = nll_sum[b * NB + i] * (1.f / NL);
  }
  __syncthreads();
  if (t == 0) {
    float mh = 0.f, ms = 0.f;
    for (int j = 0; j < NB; ++j) { mh += hid[j]; ms += sur[j]; }
    mh *= (1.f / NB); ms *= (1.f / NB);
    float vh = 0.f, vs = 0.f;
    for (int j = 0; j < NB; ++j) {
      float a = hid[j] - mh, c = sur[j] - ms;
      vh += a * a; vs += c * c;
    }
    stats[0] = mh; stats[1] = fmaxf(sqrtf(vh * (1.f / NB)), 1e-6f);
    stats[2] = ms; stats[3] = fmaxf(sqrtf(vs * (1.f / NB)), 1e-6f);
  }
  __syncthreads();
  if (t < NB)
    sc[t] = (hid[t] - stats[0]) / stats[1] + (sur[t] - stats[2]) / stats[3];
  __syncthreads();
  if (t < NB) {
    int r = 0; float v = sc[t];
    for (int j = 0; j < NB; ++j) r += (sc[j] > v) || (sc[j] == v && j < t);
    sel[t] = (r < NTOP) ? 1 : 0;
  }
  __syncthreads();
  if (t < NB && sel[t]) {
    int pos = 0;
    for (int j = 0; j < t; ++j) pos += sel[j];
    idx[b * NTOP + pos] = t;
  }
}

// ---------------------------------------------------------------------------
// Phase 3: build feats (B, 21, 2048). grid = B*NS (84), block = 256.
// ---------------------------------------------------------------------------
__global__ void k_feats(const float* __restrict__ ph, const float* __restrict__ seg_sum,
                        const int* __restrict__ idx, float* __restrict__ feats) {
  int g = blockIdx.x, b = g / NS, s = g % NS, t = threadIdx.x;
  int d = t * 4;
  v4f mean;
  const float* last_src;
  if (s < NTOP) {
    int blk = idx[b * NTOP + s];
    mean = (*(const v4f*)(seg_sum + ((size_t)(b * NB + blk)) * ND + d)) * (1.f / NL);
    last_src = ph + ((size_t)b * NT + blk * NL + (NL - 1)) * ND;
  } else if (s < NTOP + 4) {
    int m = s - NTOP;
    v4f acc = {0.f, 0.f, 0.f, 0.f};
    for (int j = 0; j < 16; ++j)
      acc += *(const v4f*)(seg_sum + ((size_t)(b * NB + m * 16 + j)) * ND + d);
    mean = acc * (1.f / 2048.f);
    last_src = ph + ((size_t)b * NT + (m + 1) * 2048 - 1) * ND;
  } else {
    v4f acc = {0.f, 0.f, 0.f, 0.f};
    for (int j = 0; j < NB; ++j)
      acc += *(const v4f*)(seg_sum + ((size_t)(b * NB + j)) * ND + d);
    mean = acc * (1.f / NT);
    last_src = ph + ((size_t)b * NT + NT - 1) * ND;
  }
  float* frow = feats + ((size_t)(b * NS + s)) * NF;
  *(v4f*)(frow + d)      = mean;
  *(v4f*)(frow + ND + d) = *(const v4f*)(last_src + d);
}

// ---------------------------------------------------------------------------
// Phase 4: summaries_raw = feats @ W_sum^T  (M=21 padded to 32, K=2048, N=1024)
// 4 waves per workgroup K-split + LDS reduce. grid = (64, B), block = 128.
// ---------------------------------------------------------------------------
__global__ void __launch_bounds__(128) k_gemm_sum(const float* __restrict__ feats,
                                                  const float* __restrict__ Wsum,
                                                  float* __restrict__ sraw) {
  __shared__ float red[KW][2][8][32];
  int nt = blockIdx.x, b = blockIdx.y;
  int tid  = threadIdx.x;
  int w    = tid >> 5;
  int lane = tid & 31;
  int nloc  = lane & 15;
  int n     = nt * 16 + nloc;
  int khalf = lane >> 4;                     // 0 or 1
  const float* A  = feats + (size_t)b * NS * NF;
  const float* Bp = Wsum + (size_t)n * NF;   // B[k][n] = W_sum[n][k]
  int m1 = 16 + nloc;
  bool m1ok = (m1 < NS);
  const float* ar0 = A + (size_t)nloc * NF;
  const float* ar1 = A + (size_t)(m1ok ? m1 : 0) * NF;
  v8f acc0 = {0.f,0.f,0.f,0.f,0.f,0.f,0.f,0.f};
  v8f acc1 = acc0;
  int kbeg = w * (NF / KW), kend = kbeg + (NF / KW);
#pragma unroll 4
  for (int k0 = kbeg; k0 < kend; k0 += 4) {
    int k = k0 + khalf * 2;
    v2f a0 = *(const v2f*)(ar0 + k);
    v2f a1 = *(const v2f*)(ar1 + k);
    if (!m1ok) { a1.x = 0.f; a1.y = 0.f; }
    v2f bb = *(const v2f*)(Bp + k);
    acc0 = wmma4(a0, bb, acc0);
    acc1 = wmma4(a1, bb, acc1);
  }
#pragma unroll
  for (int r = 0; r < 8; ++r) {
    red[w][0][r][lane] = acc0[r];
    red[w][1][r][lane] = acc1[r];
  }
  __syncthreads();
  if (w == 0) {
    int mw = khalf * 8;
    float* ocol = sraw + (size_t)b * NS * ND + (size_t)nt * 16 + nloc;
#pragma unroll
    for (int r = 0; r < 8; ++r) {
      float s0 = red[0][0][r][lane] + red[1][0][r][lane] +
                 red[2][0][r][lane] + red[3][0][r][lane];
      float s1 = red[0][1][r][lane] + red[1][1][r][lane] +
                 red[2][1][r][lane] + red[3][1][r][lane];
      int m = mw + r;
      if (m < NS)      ocol[(size_t)m * ND]      = s0;
      int m2 = 16 + m;
      if (m2 < NS)     ocol[(size_t)m2 * ND]     = s1;
    }
  }
}

// ---------------------------------------------------------------------------
// Phase 5: RMS-norm rows of (B*NS, 1024). grid = 84, block = 256.
// ---------------------------------------------------------------------------
__global__ void k_rms(const float* __restrict__ sraw, float* __restrict__ summ) {
  int g = blockIdx.x, t = threadIdx.x;
  const float* row = sraw + (size_t)g * ND;
  __shared__ float red[256];
  __shared__ float scale;
  v4f v = *(const v4f*)(row + t * 4);
  red[t] = v.x * v.x + v.y * v.y + v.z * v.z + v.w * v.w;
  __syncthreads();
  for (int off = 128; off > 0; off >>= 1) {
    if (t < off) red[t] += red[t + off];
    __syncthreads();
  }
  if (t == 0) scale = rsqrtf(red[0] * (1.f / ND) + RMS_EPS);
  __syncthreads();
  *(v4f*)(summ + (size_t)g * ND + t * 4) = v * scale;
}

// ---------------------------------------------------------------------------
// Phase 6: keys/vals = summaries @ W_k^T / W_v^T (M=21, K=1024, N=1024).
// 4 waves per workgroup K-split + LDS reduce. grid = (64, B), block = 128.
// ---------------------------------------------------------------------------
__global__ void __launch_bounds__(128) k_gemm_kv(const float* __restrict__ summ,
                                                 const float* __restrict__ Wk,
                                                 const float* __restrict__ Wv,
                                                 float* __restrict__ keys,
                                                 float* __restrict__ vals) {
  __shared__ float red[KW][4][8][32];
  int nt = blockIdx.x, b = blockIdx.y;
  int tid  = threadIdx.x;
  int w    = tid >> 5;
  int lane = tid & 31;
  int nloc  = lane & 15;
  int n     = nt * 16 + nloc;
  int khalf = lane >> 4;
  const float* A  = summ + (size_t)b * NS * ND;
  const float* Bk = Wk + (size_t)n * ND;
  const float* Bv = Wv + (size_t)n * ND;
  int m1 = 16 + nloc;
  bool m1ok = (m1 < NS);
  const float* ar0 = A + (size_t)nloc * ND;
  const float* ar1 = A + (size_t)(m1ok ? m1 : 0) * ND;
  v8f ak0 = {0.f,0.f,0.f,0.f,0.f,0.f,0.f,0.f};
  v8f ak1 = ak0, av0 = ak0, av1 = ak0;
  int kbeg = w * (ND / KW), kend = kbeg + (ND / KW);
#pragma unroll 4
  for (int k0 = kbeg; k0 < kend; k0 += 4) {
    int k = k0 + khalf * 2;
    v2f a0 = *(const v2f*)(ar0 + k);
    v2f a1 = *(const v2f*)(ar1 + k);
    if (!m1ok) { a1.x = 0.f; a1.y = 0.f; }
    v2f bk = *(const v2f*)(Bk + k);
    v2f bv = *(const v2f*)(Bv + k);
    ak0 = wmma4(a0, bk, ak0);
    ak1 = wmma4(a1, bk, ak1);
    av0 = wmma4(a0, bv, av0);
    av1 = wmma4(a1, bv, av1);
  }
#pragma unroll
  for (int r = 0; r < 8; ++r) {
    red[w][0][r][lane] = ak0[r];
    red[w][1][r][lane] = ak1[r];
    red[w][2][r][lane] = av0[r];
    red[w][3][r][lane] = av1[r];
  }
  __syncthreads();
  if (w == 0) {
    int mw = khalf * 8;
    size_t coloff = (size_t)b * NS * ND + (size_t)nt * 16 + nloc;
#pragma unroll
    for (int r = 0; r < 8; ++r) {
      float sk0 = red[0][0][r][lane] + red[1][0][r][lane] +
                  red[2][0][r][lane] + red[3][0][r][lane];
      float sk1 = red[0][1][r][lane] + red[1][1][r][lane] +
                  red[2][1][r][lane] + red[3][1][r][lane];
      float sv0 = red[0][2][r][lane] + red[1][2][r][lane] +
                  red[2][2][r][lane] + red[3][2][r][lane];
      float sv1 = red[0][3][r][lane] + red[1][3][r][lane] +
                  red[2][3][r][lane] + red[3][3][r][lane];
      int m = mw + r;
      if (m < NS) {
        keys[coloff + (size_t)m * ND] = sk0;
        vals[coloff + (size_t)m * ND] = sv0;
      }
      int m2 = 16 + m;
      if (m2 < NS) {
        keys[coloff + (size_t)m2 * ND] = sk1;
        vals[coloff + (size_t)m2 * ND] = sv1;
      }
    }
  }
}

// ---------------------------------------------------------------------------
// Phase 7: logits -> softmax(21) -> prefix. grid = B*NQ (256), block = 256.
// ---------------------------------------------------------------------------
__global__ void k_attn(const float* __restrict__ query, const float* __restrict__ keys,
                       const float* __restrict__ vals, float* __restrict__ prefix) {
  int g = blockIdx.x, b = g >> 6, q = g & 63, t = threadIdx.x;
  __shared__ float part[NS * 8];
  __shared__ float attn[NS];
  int s = t >> 3, p = t & 7;
  if (s < NS) {
    const float* kr = keys + ((size_t)(b * NS + s)) * ND + p * 128;
    const float* qr = query + (size_t)q * ND + p * 128;
    float a = 0.f;
    for (int d = 0; d < 128; ++d) a += qr[d] * kr[d];
    part[s * 8 + p] = a;
  }
  __syncthreads();
  if (t == 0) {
    float lg[NS];
    float mx = -1e30f;
    for (int j = 0; j < NS; ++j) {
      float v = 0.f;
      for (int pp = 0; pp < 8; ++pp) v += part[j * 8 + pp];
      v *= (1.f / 32.f);                    // / sqrt(D)
      lg[j] = v;
      mx = fmaxf(mx, v);
    }
    float sum = 0.f;
    for (int j = 0; j < NS; ++j) { lg[j] = expf(lg[j] - mx); sum += lg[j]; }
    float inv = 1.f / sum;
    for (int j = 0; j < NS; ++j) attn[j] = lg[j] * inv;
  }
  __syncthreads();
  int e = t * 4;
  v4f acc = {0.f, 0.f, 0.f, 0.f};
  for (int j = 0; j < NS; ++j)
    acc += attn[j] * (*(const v4f*)(vals + ((size_t)(b * NS + j)) * ND + e));
  *(v4f*)(prefix + (size_t)g * ND + e) = acc;
}

// ---------------------------------------------------------------------------
// Phase 8: out = prefix @ W_o^T (M=64, K=1024, N=1024).
// 4 waves per workgroup K-split + LDS reduce. grid = (64, B), block = 128.
// ---------------------------------------------------------------------------
__global__ void __launch_bounds__(128) k_gemm_out(const float* __restrict__ prefix,
                                                  const float* __restrict__ Wo,
                                                  float* __restrict__ out) {
  __shared__ float red[KW][4][8][32];
  int nt = blockIdx.x, b = blockIdx.y;
  int tid  = threadIdx.x;
  int w    = tid >> 5;
  int lane = tid & 31;
  int nloc  = lane & 15;
  int n     = nt * 16 + nloc;
  int khalf = lane >> 4;
  const float* A  = prefix + (size_t)b * NQ * ND;
  const float* Bp = Wo + (size_t)n * ND;       // B[k][n] = W_o[n][k]
  v8f acc[4];
#pragma unroll
  for (int mt = 0; mt < 4; ++mt) acc[mt] = (v8f){0.f,0.f,0.f,0.f,0.f,0.f,0.f,0.f};
  int kbeg = w * (ND / KW), kend = kbeg + (ND / KW);
#pragma unroll 4
  for (int k0 = kbeg; k0 < kend; k0 += 4) {
    int k = k0 + khalf * 2;
    v2f bb = *(const v2f*)(Bp + k);
#pragma unroll
    for (int mt = 0; mt < 4; ++mt) {
      v2f a = *(const v2f*)(A + (size_t)(mt * 16 + nloc) * ND + k);
      acc[mt] = wmma4(a, bb, acc[mt]);
    }
  }
#pragma unroll
  for (int mt = 0; mt < 4; ++mt)
#pragma unroll
    for (int r = 0; r < 8; ++r)
      red[w][mt][r][lane] = acc[mt][r];
  __syncthreads();
  if (w == 0) {
    int mw = khalf * 8;
#pragma unroll
    for (int mt = 0; mt < 4; ++mt) {
#pragma unroll
      for (int r = 0; r < 8; ++r) {
        float s = red[0][mt][r][lane] + red[1][mt][r][lane] +
                  red[2][mt][r][lane] + red[3][mt][r][lane];
        int m = mt * 16 + mw + r;
        out[((size_t)b * NQ + m) * ND + (size_t)nt * 16 + nloc] = s;
      }
    }
  }
}

// ---------------------------------------------------------------------------
extern "C" void kernel_launch(void* const* d_in, const int* in_sizes, int n_in,
                              void* d_out, int out_size, void* d_ws, size_t ws_size,
                              hipStream_t stream) {
  const float* ph   = (const float*)d_in[0];  // prev_hidden (4,8192,1024)
  const float* nll  = (const float*)d_in[1];  // prev_nll    (4,8192)
  const float* qry  = (const float*)d_in[2];  // query       (64,1024)
  const float* Wsum = (const float*)d_in[3];  // (1024,2048)
  const float* Wk   = (const float*)d_in[4];  // (1024,1024)
  const float* Wv   = (const float*)d_in[5];  // (1024,1024)
  const float* Wo   = (const float*)d_in[6];  // (1024,1024)
  float* out = (float*)d_out;                 // (4,64,1024)

  char* ws = (char*)d_ws;
  size_t o = 0;
  float* part    = (float*)(ws + o); o += (size_t)NBATCH * NB * QSPLIT * ND * 4; // 8 MB
  float* seg_sum = (float*)(ws + o); o += (size_t)NBATCH * NB * ND * 4;          // 1 MB
  float* nll_sum = (float*)(ws + o); o += 1024;
  int*   idx     = (int*)(ws + o);   o += 1024;
  float* feats   = (float*)(ws + o); o += (size_t)NBATCH * NS * NF * 4;
  float* sraw    = (float*)(ws + o); o += (size_t)NBATCH * NS * ND * 4;
  float* summ    = (float*)(ws + o); o += (size_t)NBATCH * NS * ND * 4;
  float* keys    = (float*)(ws + o); o += (size_t)NBATCH * NS * ND * 4;
  float* vals    = (float*)(ws + o); o += (size_t)NBATCH * NS * ND * 4;
  float* prefix  = (float*)(ws + o); o += (size_t)NBATCH * NQ * ND * 4;

  k_seg_partial<<<NBATCH * NB * QSPLIT, 256, 0, stream>>>(ph, part);
  k_seg_combine<<<NBATCH * NB, 256, 0, stream>>>(part, nll, seg_sum, nll_sum);
  k_score_topk<<<NBATCH, 256, 0, stream>>>(seg_sum, nll_sum, idx);
  k_feats<<<NBATCH * NS, 256, 0, stream>>>(ph, seg_sum, idx, feats);

  dim3 gtile(ND / 16, NBATCH);
  k_gemm_sum<<<gtile, 128, 0, stream>>>(feats, Wsum, sraw);
  k_rms<<<NBATCH * NS, 256, 0, stream>>>(sraw, summ);
  k_gemm_kv<<<gtile, 128, 0, stream>>>(summ, Wk, Wv, keys, vals);
  k_attn<<<NBATCH * NQ, 256, 0, stream>>>(qry, keys, vals, prefix);
  k_gemm_out<<<gtile, 128, 0, stream>>>(prefix, Wo, out);
}